// GNN_19164144074980
// MI455X (gfx1250) — compile-verified
//
#include <hip/hip_runtime.h>
#include <math.h>

#define NND 100000
#define NE  1600000
#define FIN 128
#define HD  64
#define NOUT 16
#define SEG_CAP 192

typedef __attribute__((ext_vector_type(2))) float v2f;
typedef __attribute__((ext_vector_type(8))) float v8f;

__device__ __forceinline__ float2 ldf2(const float* p) {
    return *reinterpret_cast<const float2*>(p);
}

// ---------------- CSR construction (once per launch) ----------------

__global__ __launch_bounds__(256) void hist_kernel(const int* __restrict__ dst,
                                                   int* __restrict__ deg) {
    int i = blockIdx.x * 256 + threadIdx.x;
    if (i < NE) atomicAdd(&deg[dst[i]], 1);
}

__global__ __launch_bounds__(1024) void scan_kernel(int* __restrict__ deg,
                                                    int* __restrict__ offs) {
    __shared__ int buf[1024];
    __shared__ int carry;
    if (threadIdx.x == 0) carry = 0;
    __syncthreads();
    for (int base = 0; base < NND; base += 1024) {
        int i = base + (int)threadIdx.x;
        int v = (i < NND) ? deg[i] : 0;
        buf[threadIdx.x] = v;
        __syncthreads();
        for (int off = 1; off < 1024; off <<= 1) {
            int t = (threadIdx.x >= (unsigned)off) ? buf[threadIdx.x - off] : 0;
            __syncthreads();
            buf[threadIdx.x] += t;
            __syncthreads();
        }
        int total = buf[1023];
        int excl  = buf[threadIdx.x] - v + carry;
        if (i < NND) { offs[i] = excl; deg[i] = excl; }  // deg becomes cursor
        __syncthreads();
        if (threadIdx.x == 0) carry += total;
        __syncthreads();
    }
    if (threadIdx.x == 0) offs[NND] = carry;  // == NE
}

__global__ __launch_bounds__(256) void scatter_kernel(const int* __restrict__ src,
                                                      const int* __restrict__ dst,
                                                      int* __restrict__ cursor,
                                                      int* __restrict__ ssort) {
    int i = blockIdx.x * 256 + threadIdx.x;
    if (i < NE) {
        int p = atomicAdd(&cursor[dst[i]], 1);
        ssort[p] = src[i];
    }
}

// Sort each dst-segment by src value -> deterministic fp accumulation order.
__global__ __launch_bounds__(256) void segsort_kernel(int* __restrict__ ssort,
                                                      const int* __restrict__ offs) {
    __shared__ int sbuf[8][SEG_CAP];
    const int wave = threadIdx.x >> 5;
    const int lane = threadIdx.x & 31;
    const int node = blockIdx.x * 8 + wave;
    if (node >= NND) return;
    const int beg = offs[node];
    const int len = offs[node + 1] - beg;
    if (len <= 1 || len > SEG_CAP) return;
    volatile int* b = sbuf[wave];
    for (int i = lane; i < len; i += 32) b[i] = ssort[beg + i];
    __builtin_amdgcn_wave_barrier();
    for (int it = 0; it < len; ++it) {
        const int p = it & 1;
        for (int j = p + 2 * lane; j + 1 < len; j += 64) {
            int x = b[j], y = b[j + 1];
            if (x > y) { b[j] = y; b[j + 1] = x; }
        }
        __builtin_amdgcn_wave_barrier();
    }
    for (int i = lane; i < len; i += 32) ssort[beg + i] = b[i];
}

// ---------------- fused 4-gate GEMM: {k,q,v,s} = X @ W[g] ----------------
// Block = 128 threads = 4 waves; wave g computes gate g over a shared
// 32-row X tile in LDS. fp32 WMMA 16x16x4: per wave 2 mtiles x 4 ntiles.

template <int KD>
__global__ __launch_bounds__(128) void gemm4_kernel(const float* __restrict__ X,
                                                    const float* __restrict__ W,
                                                    float* __restrict__ ko,
                                                    float* __restrict__ qo,
                                                    float* __restrict__ vo,
                                                    float* __restrict__ so) {
    __shared__ __align__(16) float xt[32 * KD];
    const int tid  = threadIdx.x;
    const int wave = tid >> 5;
    const int lane = tid & 31;
    const int row0 = blockIdx.x * 32;  // N % 32 == 0

    // X rows are contiguous: copy 32*KD floats via float4.
    {
        const float4* s4 = reinterpret_cast<const float4*>(X + (size_t)row0 * KD);
        float4* d4 = reinterpret_cast<float4*>(xt);
        constexpr int NV4 = 32 * KD / 4;
        #pragma unroll
        for (int i = tid; i < NV4; i += 128) d4[i] = s4[i];
    }
    __syncthreads();

    const float* Wg = W + (size_t)wave * KD * HD;  // [KD][HD] for this gate
    const int mlan = lane & 15;
    const int ksel = (lane >> 4) << 1;  // 0 or 2

    v8f c[2][4];
    const v8f vzero = {0.f, 0.f, 0.f, 0.f, 0.f, 0.f, 0.f, 0.f};
    #pragma unroll
    for (int mt = 0; mt < 2; ++mt)
        #pragma unroll
        for (int nt = 0; nt < 4; ++nt) c[mt][nt] = vzero;

    #pragma unroll 4
    for (int kt = 0; kt < KD; kt += 4) {
        // A fragments (16x4): lane m, K = kt+ksel, kt+ksel+1
        const float* xr0 = &xt[(size_t)mlan * KD + kt + ksel];
        const float* xr1 = &xt[(size_t)(16 + mlan) * KD + kt + ksel];
        v2f a0 = {xr0[0], xr0[1]};
        v2f a1 = {xr1[0], xr1[1]};
        // B fragments (4x16): rows kt+ksel, kt+ksel+1, col = nt*16+mlan
        const float* wr = Wg + (size_t)(kt + ksel) * HD + mlan;
        #pragma unroll
        for (int nt = 0; nt < 4; ++nt) {
            v2f b = {wr[nt * 16], wr[HD + nt * 16]};
            c[0][nt] = __builtin_amdgcn_wmma_f32_16x16x4_f32(
                false, a0, false, b, (short)0, c[0][nt], false, false);
            c[1][nt] = __builtin_amdgcn_wmma_f32_16x16x4_f32(
                false, a1, false, b, (short)0, c[1][nt], false, false);
        }
    }

    float* outp = (wave == 0) ? ko : (wave == 1) ? qo : (wave == 2) ? vo : so;
    const int rbump = (lane < 16) ? 0 : 8;
    #pragma unroll
    for (int mt = 0; mt < 2; ++mt)
        #pragma unroll
        for (int r = 0; r < 8; ++r) {
            const int row = row0 + mt * 16 + rbump + r;
            #pragma unroll
            for (int nt = 0; nt < 4; ++nt)
                outp[(size_t)row * HD + nt * 16 + mlan] = c[mt][nt][r];
        }
}

// ---------------- fused edge aggregation + skip + bias + leaky_relu ---------
// One wave per node; lane handles 2 of the 64 hidden dims (float2).
// Gathers of q/v are L2-resident (25.6 MB arrays, 192 MB L2).

__global__ __launch_bounds__(256) void edge_agg_kernel(const float* __restrict__ kb,
                                                       const float* __restrict__ qb,
                                                       const float* __restrict__ vb,
                                                       const float* __restrict__ sb,
                                                       const float* __restrict__ bias,
                                                       const int* __restrict__ srcs,
                                                       const int* __restrict__ offs,
                                                       float* __restrict__ hout) {
    const int wave = threadIdx.x >> 5;
    const int lane = threadIdx.x & 31;
    const int node = blockIdx.x * 8 + wave;
    if (node >= NND) return;
    const int d0 = lane << 1;
    const float2 kk = ldf2(kb + (size_t)node * HD + d0);
    float ax = 0.f, ay = 0.f;
    const int beg = offs[node];
    const int end = offs[node + 1];
    int s = (beg < end) ? srcs[beg] : 0;
    for (int e = beg; e < end; ++e) {
        const int snext = (e + 1 < end) ? srcs[e + 1] : 0;
        const float2 q2 = ldf2(qb + (size_t)s * HD + d0);
        const float2 v2 = ldf2(vb + (size_t)s * HD + d0);
        const float gx = 1.f / (1.f + __expf(-(kk.x + q2.x)));
        const float gy = 1.f / (1.f + __expf(-(kk.y + q2.y)));
        ax = __fmaf_rn(gx, v2.x, ax);
        ay = __fmaf_rn(gy, v2.y, ay);
        s = snext;
    }
    const float2 sv = ldf2(sb + (size_t)node * HD + d0);
    float ox = sv.x + ax + bias[d0];
    float oy = sv.y + ay + bias[d0 + 1];
    ox = (ox > 0.f) ? ox : 0.01f * ox;
    oy = (oy > 0.f) ? oy : 0.01f * oy;
    float2 o = make_float2(ox, oy);
    *reinterpret_cast<float2*>(hout + (size_t)node * HD + d0) = o;
}

// ---------------- final projection: out = H @ W_out + b_out ----------------

__global__ __launch_bounds__(256) void out_gemm_kernel(const float* __restrict__ Hm,
                                                       const float* __restrict__ Wo,
                                                       const float* __restrict__ bo,
                                                       float* __restrict__ out) {
    const int wave = threadIdx.x >> 5;
    const int lane = threadIdx.x & 31;
    const int row0 = blockIdx.x * 128 + wave * 16;
    if (row0 >= NND) return;  // N % 16 == 0 -> tiles fully valid or fully out
    const int mlan = lane & 15;
    const int ksel = (lane >> 4) << 1;
    v8f c = {0.f, 0.f, 0.f, 0.f, 0.f, 0.f, 0.f, 0.f};
    #pragma unroll
    for (int kt = 0; kt < HD; kt += 4) {
        const float* xr = Hm + (size_t)(row0 + mlan) * HD + kt + ksel;
        v2f a = {xr[0], xr[1]};
        const float* wr = Wo + (size_t)(kt + ksel) * NOUT + mlan;
        v2f b = {wr[0], wr[NOUT]};
        c = __builtin_amdgcn_wmma_f32_16x16x4_f32(false, a, false, b, (short)0, c,
                                                  false, false);
    }
    const int rbump = (lane < 16) ? 0 : 8;
    const float bias = bo[mlan];
    #pragma unroll
    for (int r = 0; r < 8; ++r)
        out[(size_t)(row0 + rbump + r) * NOUT + mlan] = c[r] + bias;
}

// ---------------- host orchestration ----------------

extern "C" void kernel_launch(void* const* d_in, const int* in_sizes, int n_in,
                              void* d_out, int out_size, void* d_ws, size_t ws_size,
                              hipStream_t stream) {
    const float* x     = (const float*)d_in[0];
    const int*   ei    = (const int*)d_in[1];
    const float* W_in  = (const float*)d_in[2];
    const float* b_in  = (const float*)d_in[3];
    const float* W_h   = (const float*)d_in[4];
    const float* b_h   = (const float*)d_in[5];
    const float* W_out = (const float*)d_in[6];
    const float* b_out = (const float*)d_in[7];
    const int* src = ei;
    const int* dst = ei + NE;

    const size_t NH = (size_t)NND * HD;
    char* ws = (char*)d_ws;
    float* kb = (float*)ws; ws += NH * sizeof(float);
    float* qb = (float*)ws; ws += NH * sizeof(float);
    float* vb = (float*)ws; ws += NH * sizeof(float);
    float* sb = (float*)ws; ws += NH * sizeof(float);
    float* hb = (float*)ws; ws += NH * sizeof(float);
    int* deg   = (int*)ws; ws += (size_t)NND * sizeof(int);       // later: cursor
    int* offs  = (int*)ws; ws += (size_t)(NND + 1) * sizeof(int);
    int* ssort = (int*)ws; ws += (size_t)NE * sizeof(int);

    // Build dst-sorted CSR of src indices (deterministic after segsort).
    (void)hipMemsetAsync(deg, 0, (size_t)NND * sizeof(int), stream);
    hist_kernel<<<NE / 256, 256, 0, stream>>>(dst, deg);
    scan_kernel<<<1, 1024, 0, stream>>>(deg, offs);
    scatter_kernel<<<NE / 256, 256, 0, stream>>>(src, dst, deg, ssort);
    segsort_kernel<<<NND / 8, 256, 0, stream>>>(ssort, offs);

    // Input conv: 128 -> 64.
    gemm4_kernel<FIN><<<NND / 32, 128, 0, stream>>>(x, W_in, kb, qb, vb, sb);
    edge_agg_kernel<<<NND / 8, 256, 0, stream>>>(kb, qb, vb, sb, b_in, ssort, offs, hb);

    // 3 repeats of 4 hidden convs (weight sharing).
    for (int rep = 0; rep < 3; ++rep) {
        for (int l = 0; l < 4; ++l) {
            const float* Wl = W_h + (size_t)l * 4 * HD * HD;
            const float* bl = b_h + (size_t)l * HD;
            gemm4_kernel<HD><<<NND / 32, 128, 0, stream>>>(hb, Wl, kb, qb, vb, sb);
            edge_agg_kernel<<<NND / 8, 256, 0, stream>>>(kb, qb, vb, sb, bl, ssort, offs, hb);
        }
    }

    // Final projection 64 -> 16.
    out_gemm_kernel<<<(NND + 127) / 128, 256, 0, stream>>>(hb, W_out, b_out,
                                                           (float*)d_out);
}